// Graph_Layer_38328288149929
// MI455X (gfx1250) — compile-verified
//
#include <hip/hip_runtime.h>
#include <float.h>

#define BB 8
#define CC 256
#define TT 2048
#define KNB 10      // K neighbors
#define HALF1 5
#define HALF2 5
#define BIGV 3.0e38f

typedef __attribute__((ext_vector_type(2))) float v2f;
typedef __attribute__((ext_vector_type(8))) float v8f;
typedef __attribute__((ext_vector_type(4))) unsigned int u32x4;
typedef __attribute__((ext_vector_type(8))) int i32x8;
typedef __attribute__((ext_vector_type(4))) int i32x4;

#if defined(__has_builtin)
#  if __has_builtin(__builtin_amdgcn_tensor_load_to_lds) && __has_builtin(__builtin_amdgcn_s_wait_tensorcnt)
#    define HAVE_TDM 1
#  endif
#endif
#ifndef HAVE_TDM
#  define HAVE_TDM 0
#endif

// ---------------------------------------------------------------------------
// sq[b*T+t] = sum_c x[b,c,t]^2
// ---------------------------------------------------------------------------
__global__ void sq_kernel(const float* __restrict__ x, float* __restrict__ sq) {
    int n = blockIdx.x * blockDim.x + threadIdx.x;      // 0 .. B*T-1
    int b = n / TT, t = n % TT;
    const float* xp = x + (size_t)b * CC * TT + t;
    float s = 0.f;
    for (int c = 0; c < CC; ++c) {
        float v = xp[(size_t)c * TT];
        s += v * v;
    }
    sq[n] = s;
}

// ---------------------------------------------------------------------------
// Conv1d(k=3,pad=1) as GEMM: M=o(256), N=t(2048), K=kk*256+ci (768).
// One wave -> one 16x16 tile. Output stored transposed: cout_tr[(b*T+t)*C + o].
// Boundary handled by clamped-address load + select (no exec-mask branches).
// ---------------------------------------------------------------------------
__global__ void conv_wmma_kernel(const float* __restrict__ x,
                                 const float* __restrict__ Wc,
                                 const float* __restrict__ bc,
                                 float* __restrict__ cout_tr) {
    int wave = (blockIdx.x * blockDim.x + threadIdx.x) >> 5;
    int lane = threadIdx.x & 31;
    int tt = wave & 127;            // t-tile 0..127
    int ot = (wave >> 7) & 15;      // o-tile 0..15
    int b  = wave >> 11;            // batch 0..7
    int t0 = tt * 16, o0 = ot * 16;
    int lo = lane & 15, hi = lane >> 4;

    const float* xb = x + (size_t)b * CC * TT;
    v8f acc = {0.f,0.f,0.f,0.f,0.f,0.f,0.f,0.f};

    for (int k0 = 0; k0 < 768; k0 += 4) {
        int ka = k0 + 2 * hi;
        v2f a, bv;
        {   // A[m,k] = W_conv[o0+m, ci, kk], k = kk*256+ci
            int k = ka, ci = k & 255, kk = k >> 8;
            a.x = Wc[(size_t)(o0 + lo) * 768 + ci * 3 + kk];
            k = ka + 1; ci = k & 255; kk = k >> 8;
            a.y = Wc[(size_t)(o0 + lo) * 768 + ci * 3 + kk];
        }
        {   // B[k,n] = x[b, ci, t0+n+kk-1] with zero pad (branch-free)
            int k = ka, ci = k & 255, kk = k >> 8;
            int tp = t0 + lo + kk - 1;
            int tpc = tp < 0 ? 0 : (tp > TT - 1 ? TT - 1 : tp);
            float xv = xb[(size_t)ci * TT + tpc];
            bv.x = (tp == tpc) ? xv : 0.f;
            k = ka + 1; ci = k & 255; kk = k >> 8;
            tp = t0 + lo + kk - 1;
            tpc = tp < 0 ? 0 : (tp > TT - 1 ? TT - 1 : tp);
            xv = xb[(size_t)ci * TT + tpc];
            bv.y = (tp == tpc) ? xv : 0.f;
        }
        acc = __builtin_amdgcn_wmma_f32_16x16x4_f32(false, a, false, bv,
                                                    (short)0, acc, false, false);
    }
    // C/D layout: lane lo holds N=lo, M = hi*8 + r -> 8 contiguous o per lane
    int obase = o0 + hi * 8;
    float* dst = cout_tr + ((size_t)b * TT + t0 + lo) * CC + obase;
    #pragma unroll
    for (int r = 0; r < 8; ++r) dst[r] = acc[r] + bc[obase + r];
}

// ---------------------------------------------------------------------------
// H1 = X^T W1^T, H2 = X^T W2^T  (W1 = W_mlp[:, :256], W2 = W_mlp[:, 256:])
// Stored as H[(b*T+t)*C + o]. One wave per 16x16 tile; sel picks H1/H2.
// ---------------------------------------------------------------------------
__global__ void mlp_h_wmma_kernel(const float* __restrict__ x,
                                  const float* __restrict__ Wm,
                                  float* __restrict__ h1,
                                  float* __restrict__ h2) {
    int wave = (blockIdx.x * blockDim.x + threadIdx.x) >> 5;
    int lane = threadIdx.x & 31;
    int tt  = wave & 127;
    int ot  = (wave >> 7) & 15;
    int b   = (wave >> 11) & 7;
    int sel = wave >> 14;                 // 0 -> H1, 1 -> H2
    int t0 = tt * 16, o0 = ot * 16;
    int lo = lane & 15, hi = lane >> 4;
    int off = sel ? CC : 0;
    float* H = sel ? h2 : h1;

    const float* xb = x + (size_t)b * CC * TT;
    v8f acc = {0.f,0.f,0.f,0.f,0.f,0.f,0.f,0.f};

    for (int k0 = 0; k0 < CC; k0 += 4) {
        int ka = k0 + 2 * hi;
        v2f a, bv;
        a.x  = Wm[(size_t)(o0 + lo) * 512 + off + ka];
        a.y  = Wm[(size_t)(o0 + lo) * 512 + off + ka + 1];
        bv.x = xb[(size_t)ka * TT + t0 + lo];
        bv.y = xb[(size_t)(ka + 1) * TT + t0 + lo];
        acc = __builtin_amdgcn_wmma_f32_16x16x4_f32(false, a, false, bv,
                                                    (short)0, acc, false, false);
    }
    int obase = o0 + hi * 8;
    float* dst = H + ((size_t)b * TT + t0 + lo) * CC + obase;
    #pragma unroll
    for (int r = 0; r < 8; ++r) dst[r] = acc[r];
}

// ---------------------------------------------------------------------------
// KNN: one block (4 waves) per (b, 16-row i-tile).
// The i-tile x[b, 0:256, i0:i0+16] (16 KB) is staged into LDS once via the
// Tensor Data Mover (2D tile: 16 contiguous t-elems x 256 channel rows,
// row stride T), synchronized with s_wait_tensorcnt + block barrier.
// Each wave computes a 16x16 gram tile via WMMA (A from LDS, B from L2);
// score = sq_j - 2*gram (sq_i is a row-constant shift: order-invariant).
// 16 threads keep per-row sorted lists:
//   listA: top-10 unmasked; listB: top-15 region-masked.
// tail = cond ? (listB minus listA[0:5])[0:5] : listA[5:10].
// ---------------------------------------------------------------------------
__global__ void knn_kernel(const float* __restrict__ x,
                           const float* __restrict__ sq,
                           const int* __restrict__ num_frms,
                           int* __restrict__ neigh) {
    __shared__ float xi[CC * 16];      // 16 KB i-tile, xi[c*16 + t]
    __shared__ float difbuf[4 * 256];
    __shared__ float lAv[16][KNB];
    __shared__ int   lAi[16][KNB];
    __shared__ float lBv[16][15];
    __shared__ int   lBi[16][15];
    __shared__ float s_thr;
    __shared__ int   s_cond;

    int b  = blockIdx.x >> 7;
    int i0 = (blockIdx.x & 127) * 16;
    int tid = threadIdx.x;
    int lane = tid & 31;
    int w = tid >> 5;
    int lo = lane & 15, hi = lane >> 4;

    const float* xb = x + (size_t)b * CC * TT;

    if (tid == 0) {
        float nf = (float)num_frms[b];
        float ratio = 2048.0f / (float)TT;            // == 1 here
        s_thr  = (nf + 20.0f) / ratio;
        s_cond = (nf <= 0.4f * 2048.0f) ? 1 : 0;
    }
    if (tid < 16) {
        for (int p = 0; p < KNB; ++p) { lAv[tid][p] = BIGV; lAi[tid][p] = i0 + tid; }
        for (int p = 0; p < 15;  ++p) { lBv[tid][p] = BIGV; lBi[tid][p] = i0 + tid; }
    }

#if HAVE_TDM
    if (w == 0) {
        // ---- Tensor DMA descriptor (D#) : 2D tile, 4-byte elements ----
        unsigned long long ga = (unsigned long long)(uintptr_t)(xb + i0);
        unsigned int lds_off = (unsigned int)(uintptr_t)(&xi[0]); // low 32b = LDS offset
        u32x4 g0;
        g0[0] = (unsigned int)__builtin_amdgcn_readfirstlane(1);          // count=1, user desc
        g0[1] = (unsigned int)__builtin_amdgcn_readfirstlane((int)lds_off);
        g0[2] = (unsigned int)__builtin_amdgcn_readfirstlane((int)(ga & 0xffffffffu));
        g0[3] = (unsigned int)__builtin_amdgcn_readfirstlane(
                    (int)(((ga >> 32) & 0x1ffffffu) | (2u << 30)));       // addr[56:32] | type=2
        i32x8 g1;
        g1[0] = __builtin_amdgcn_readfirstlane(2 << 16);                  // data_size=4B
        g1[1] = __builtin_amdgcn_readfirstlane((int)(((unsigned)TT & 0xffffu) << 16)); // tensor_dim0 lo
        g1[2] = __builtin_amdgcn_readfirstlane((int)(((unsigned)TT >> 16) |
                                                     (((unsigned)CC & 0xffffu) << 16))); // dim0 hi | dim1 lo
        g1[3] = __builtin_amdgcn_readfirstlane((int)(((unsigned)CC >> 16) |
                                                     (16u << 16)));       // dim1 hi | tile_dim0=16
        g1[4] = __builtin_amdgcn_readfirstlane((int)(unsigned)CC);        // tile_dim1=256, tile_dim2=0
        g1[5] = __builtin_amdgcn_readfirstlane((int)(unsigned)TT);        // tensor_dim0_stride[31:0]=T
        g1[6] = __builtin_amdgcn_readfirstlane(0);                        // stride hi | dim1_stride lo
        g1[7] = __builtin_amdgcn_readfirstlane(0);
        i32x4 z4;
        z4[0] = 0; z4[1] = 0; z4[2] = 0; z4[3] = 0;
        i32x8 z8;
        z8[0] = 0; z8[1] = 0; z8[2] = 0; z8[3] = 0;
        z8[4] = 0; z8[5] = 0; z8[6] = 0; z8[7] = 0;
        // 6-arg form (clang-23 / therock-10.0 headers)
        __builtin_amdgcn_tensor_load_to_lds(g0, g1, z4, z4, z8, 0);
        __builtin_amdgcn_s_wait_tensorcnt(0);
    }
#else
    // fallback: cooperative global -> LDS fill of the i-tile
    for (int c = tid >> 4; c < CC; c += 8)
        xi[c * 16 + (tid & 15)] = xb[(size_t)c * TT + i0 + (tid & 15)];
#endif
    __syncthreads();
    float thr = s_thr;

    for (int iter = 0; iter < 32; ++iter) {           // 32 iters * 4 waves * 16 cols = 2048
        int j0 = (iter * 4 + w) * 16;
        v8f acc = {0.f,0.f,0.f,0.f,0.f,0.f,0.f,0.f};
        for (int k0 = 0; k0 < CC; k0 += 4) {
            int ka = k0 + 2 * hi;
            v2f a, bv;
            a.x  = xi[ka * 16 + lo];                  // A from LDS (TDM-staged)
            a.y  = xi[(ka + 1) * 16 + lo];
            bv.x = xb[(size_t)ka * TT + j0 + lo];     // B from global (L2-resident)
            bv.y = xb[(size_t)(ka + 1) * TT + j0 + lo];
            acc = __builtin_amdgcn_wmma_f32_16x16x4_f32(false, a, false, bv,
                                                        (short)0, acc, false, false);
        }
        float sqj = sq[b * TT + j0 + lo];
        int mbase = hi * 8;
        #pragma unroll
        for (int r = 0; r < 8; ++r)
            difbuf[w * 256 + (mbase + r) * 16 + lo] = sqj - 2.0f * acc[r];
        __syncthreads();

        if (tid < 16) {
            int m = tid;
            bool rowmask = ((float)(i0 + m) >= thr);
            for (int ww = 0; ww < 4; ++ww) {          // ascending j for stable ties
                int jb = (iter * 4 + ww) * 16;
                for (int n = 0; n < 16; ++n) {
                    float v = difbuf[ww * 256 + m * 16 + n];
                    int j = jb + n;
                    if (v < lAv[m][KNB - 1]) {        // strict < : lowest index wins ties
                        int p = KNB - 1;
                        while (p > 0 && v < lAv[m][p - 1]) {
                            lAv[m][p] = lAv[m][p - 1]; lAi[m][p] = lAi[m][p - 1]; --p;
                        }
                        lAv[m][p] = v; lAi[m][p] = j;
                    }
                    float vm = (rowmask && ((float)j >= thr)) ? BIGV : v;
                    if (vm < lBv[m][14]) {
                        int p = 14;
                        while (p > 0 && vm < lBv[m][p - 1]) {
                            lBv[m][p] = lBv[m][p - 1]; lBi[m][p] = lBi[m][p - 1]; --p;
                        }
                        lBv[m][p] = vm; lBi[m][p] = j;
                    }
                }
            }
        }
        __syncthreads();
    }

    if (tid < 16) {
        int m = tid;
        size_t base = ((size_t)b * TT + i0 + m) * KNB;
        int a0 = lAi[m][0], a1 = lAi[m][1], a2 = lAi[m][2], a3 = lAi[m][3], a4 = lAi[m][4];
        neigh[base + 0] = b * TT + a0;
        neigh[base + 1] = b * TT + a1;
        neigh[base + 2] = b * TT + a2;
        neigh[base + 3] = b * TT + a3;
        neigh[base + 4] = b * TT + a4;
        if (s_cond) {
            int cnt = 0;
            for (int p = 0; p < 15 && cnt < HALF2; ++p) {
                int j = lBi[m][p];
                if (j != a0 && j != a1 && j != a2 && j != a3 && j != a4) {
                    neigh[base + HALF1 + cnt] = b * TT + j; ++cnt;
                }
            }
            while (cnt < HALF2) { neigh[base + HALF1 + cnt] = b * TT + i0 + m; ++cnt; }
        } else {
            for (int p = HALF1; p < KNB; ++p) neigh[base + p] = b * TT + lAi[m][p];
        }
    }
}

// ---------------------------------------------------------------------------
// out[b,c,t2] = max over t in {2t2, 2t2+1} of
//     relu( cout_tr[n,c] + h2[n,c] + b_mlp[c] + max_k h1[neigh[n,k], c] )
// ---------------------------------------------------------------------------
__global__ void final_kernel(const float* __restrict__ cout_tr,
                             const float* __restrict__ h1,
                             const float* __restrict__ h2,
                             const float* __restrict__ bm,
                             const int* __restrict__ neigh,
                             float* __restrict__ out) {
    int idx = blockIdx.x * blockDim.x + threadIdx.x;   // B*C*(T/2)
    if (idx >= BB * CC * (TT / 2)) return;
    int t2 = idx % (TT / 2);
    int c  = (idx / (TT / 2)) % CC;
    int b  = idx / ((TT / 2) * CC);
    float bmv = bm[c];
    float best = 0.f;                                   // == max of relu'd values
    #pragma unroll
    for (int d = 0; d < 2; ++d) {
        int t = 2 * t2 + d;
        size_t n = (size_t)b * TT + t;
        float mx = -FLT_MAX;
        for (int p = 0; p < KNB; ++p) {
            int nb = neigh[n * KNB + p];
            mx = fmaxf(mx, h1[(size_t)nb * CC + c]);
        }
        float v = cout_tr[n * CC + c] + h2[n * CC + c] + bmv + mx;
        best = fmaxf(best, fmaxf(v, 0.f));
    }
    out[idx] = best;
}

// ---------------------------------------------------------------------------
extern "C" void kernel_launch(void* const* d_in, const int* in_sizes, int n_in,
                              void* d_out, int out_size, void* d_ws, size_t ws_size,
                              hipStream_t stream) {
    (void)in_sizes; (void)n_in; (void)out_size; (void)ws_size;
    const float* x  = (const float*)d_in[0];
    const int*   nf = (const int*)d_in[1];
    const float* Wc = (const float*)d_in[2];
    const float* bc = (const float*)d_in[3];
    const float* Wm = (const float*)d_in[4];
    const float* bm = (const float*)d_in[5];
    float* out = (float*)d_out;

    // workspace layout (bytes): sq | cout_tr | h1 | h2 | neigh  (~48.8 MB)
    char* ws = (char*)d_ws;
    float* sq      = (float*)ws;                                   // B*T
    float* cout_tr = (float*)(ws + (size_t)BB * TT * sizeof(float));
    float* h1      = cout_tr + (size_t)BB * TT * CC;
    float* h2      = h1      + (size_t)BB * TT * CC;
    int*   neigh   = (int*)(h2 + (size_t)BB * TT * CC);

    sq_kernel       <<<(BB * TT) / 256, 256, 0, stream>>>(x, sq);
    conv_wmma_kernel<<<(BB * 16 * 128) / 4, 128, 0, stream>>>(x, Wc, bc, cout_tr);
    mlp_h_wmma_kernel<<<(2 * BB * 16 * 128) / 4, 128, 0, stream>>>(x, Wm, h1, h2);
    knn_kernel      <<<BB * (TT / 16), 128, 0, stream>>>(x, sq, nf, neigh);
    final_kernel    <<<(BB * CC * (TT / 2)) / 256, 256, 0, stream>>>(cout_tr, h1, h2, bm, neigh, out);
}